// GaussianVoxelizer_27273042330210
// MI455X (gfx1250) — compile-verified
//
#include <hip/hip_runtime.h>
#include <math.h>

typedef float v2f __attribute__((ext_vector_type(2)));
typedef float v8f __attribute__((ext_vector_type(8)));

#define NVOX3  (256*256*256)
#define KWIN   12
#define KC3    (KWIN*KWIN*KWIN)   /* 1728 */
#define NTILES (KC3/16)           /* 108  */
#define LOG2E  1.4426950408889634f
#define P_MIN  (-7.9943534f)      /* log2(1/255): alpha cutoff in exp2 domain */

// ---------------------------------------------------------------------------
// Zero the 256^3 field region of d_out (harness poisons it with 0xAA).
// ---------------------------------------------------------------------------
__global__ void __launch_bounds__(256) gv_zero_fields(float* __restrict__ out) {
    float4* o4 = (float4*)out;
    const size_t n4 = NVOX3 / 4;
    for (size_t j = (size_t)blockIdx.x * blockDim.x + threadIdx.x; j < n4;
         j += (size_t)gridDim.x * blockDim.x) {
        o4[j] = make_float4(0.f, 0.f, 0.f, 0.f);
    }
}

// ---------------------------------------------------------------------------
// Splat kernel: one 256-thread block (8 wave32) per 16 gaussians.
// power[voxel, gaussian] = F[1728x12] @ K[12x16] via v_wmma_f32_16x16x4_f32.
// Opacity and log2(e) folded into K so alpha = v_exp_f32(power) directly.
// ---------------------------------------------------------------------------
__global__ void __launch_bounds__(256) gv_splat(
    const float* __restrict__ means,   // [N,3]
    const float* __restrict__ opac,    // [N,1]
    const float* __restrict__ scales,  // [N,3]
    const float* __restrict__ rots,    // [N,4]
    float* __restrict__ out,           // fields [256^3] then radii [N,3]
    int N)
{
    __shared__ float sCoef[16][12];    // log2-domain polynomial coefficients
    __shared__ int   sI0[16][3];       // window start voxel
    __shared__ int   sPack[KC3];       // ox<<16 | oy<<8 | oz  (== flat delta)

    const float h   = 2.0f / 256.0f;   // voxel size (uniform on all axes)
    const float org = -1.0f;           // grid corner (uniform on all axes)

    const int tid = threadIdx.x;

    // ---- window-offset pack table: flat delta == ox<<16 | oy<<8 | oz ----
    for (int v = tid; v < KC3; v += 256) {
        const int ox = v / 144;
        const int rm = v - ox * 144;
        const int oy = rm / 12;
        const int oz = rm - oy * 12;
        sPack[v] = (ox << 16) | (oy << 8) | oz;
    }

    // ---- per-gaussian setup: conic, coefficients, radii (threads 0..15) ----
    if (tid < 16) {
        const int g = blockIdx.x * 16 + tid;
        float k[12];
        #pragma unroll
        for (int j = 0; j < 12; ++j) k[j] = 0.0f;
        k[0] = -1e30f;                       // padded lanes never pass cutoff
        int i0x = 0, i0y = 0, i0z = 0;

        if (g < N) {
            const float mx = means[g*3+0], my = means[g*3+1], mz = means[g*3+2];
            const float op = opac[g];
            const float sx = scales[g*3+0], sy = scales[g*3+1], sz = scales[g*3+2];
            float qr = rots[g*4+0], qx = rots[g*4+1], qy = rots[g*4+2], qz = rots[g*4+3];
            const float qn = rsqrtf(qr*qr + qx*qx + qy*qy + qz*qz);
            qr *= qn; qx *= qn; qy *= qn; qz *= qn;

            const float R00 = 1.f - 2.f*(qy*qy + qz*qz);
            const float R01 = 2.f*(qx*qy - qr*qz);
            const float R02 = 2.f*(qx*qz + qr*qy);
            const float R10 = 2.f*(qx*qy + qr*qz);
            const float R11 = 1.f - 2.f*(qx*qx + qz*qz);
            const float R12 = 2.f*(qy*qz - qr*qx);
            const float R20 = 2.f*(qx*qz - qr*qy);
            const float R21 = 2.f*(qy*qz + qr*qx);
            const float R22 = 1.f - 2.f*(qx*qx + qy*qy);

            // M = R * diag(s); cov = M M^T
            const float M00 = R00*sx, M01 = R01*sy, M02 = R02*sz;
            const float M10 = R10*sx, M11 = R11*sy, M12 = R12*sz;
            const float M20 = R20*sx, M21 = R21*sy, M22 = R22*sz;
            const float c00 = M00*M00 + M01*M01 + M02*M02;
            const float c01 = M00*M10 + M01*M11 + M02*M12;
            const float c02 = M00*M20 + M01*M21 + M02*M22;
            const float c11 = M10*M10 + M11*M11 + M12*M12;
            const float c12 = M10*M20 + M11*M21 + M12*M22;
            const float c22 = M20*M20 + M21*M21 + M22*M22;

            // conic = inv(cov)
            const float det = c00*(c11*c22 - c12*c12)
                            - c01*(c01*c22 - c12*c02)
                            + c02*(c01*c12 - c11*c02);
            const float id  = 1.0f / det;
            const float a00 = (c11*c22 - c12*c12) * id;
            const float a01 = (c02*c12 - c01*c22) * id;
            const float a02 = (c01*c12 - c02*c11) * id;
            const float a11 = (c00*c22 - c02*c02) * id;
            const float a12 = (c01*c02 - c00*c12) * id;
            const float a22 = (c00*c11 - c01*c01) * id;

            const int icx = (int)floorf((mx - org) / h);
            const int icy = (int)floorf((my - org) / h);
            const int icz = (int)floorf((mz - org) / h);
            i0x = icx - KWIN/2; i0y = icy - KWIN/2; i0z = icz - KWIN/2;
            const float d0x = org + ((float)i0x + 0.5f)*h - mx;
            const float d0y = org + ((float)i0y + 0.5f)*h - my;
            const float d0z = org + ((float)i0z + 0.5f)*h - mz;

            // log2-domain coefficients: alpha = exp2(k . f(o))
            k[0] = LOG2E * (-0.5f*(a00*d0x*d0x + a11*d0y*d0y + a22*d0z*d0z)
                            - (a01*d0x*d0y + a02*d0x*d0z + a12*d0y*d0z))
                   + log2f(op);
            k[1] = LOG2E * (-h * (a00*d0x + a01*d0y + a02*d0z));
            k[2] = LOG2E * (-h * (a01*d0x + a11*d0y + a12*d0z));
            k[3] = LOG2E * (-h * (a02*d0x + a12*d0y + a22*d0z));
            k[4] = LOG2E * (-0.5f*h*h*a00);
            k[5] = LOG2E * (-0.5f*h*h*a11);
            k[6] = LOG2E * (-0.5f*h*h*a22);
            k[7] = LOG2E * (-h*h*a01);
            k[8] = LOG2E * (-h*h*a02);
            k[9] = LOG2E * (-h*h*a12);

            // radii = ceil(3*sigma/h) per axis
            out[(size_t)NVOX3 + (size_t)g*3 + 0] = ceilf(3.0f*sqrtf(c00)/h);
            out[(size_t)NVOX3 + (size_t)g*3 + 1] = ceilf(3.0f*sqrtf(c11)/h);
            out[(size_t)NVOX3 + (size_t)g*3 + 2] = ceilf(3.0f*sqrtf(c22)/h);
        }

        #pragma unroll
        for (int j = 0; j < 12; ++j) sCoef[tid][j] = k[j];
        sI0[tid][0] = i0x; sI0[tid][1] = i0y; sI0[tid][2] = i0z;
    }
    __syncthreads();

    // ---- per-lane column state (gaussian n), hoisted out of tile loop ----
    const int lane = tid & 31;
    const int n    = lane & 15;        // gaussian column of D this lane holds
    const int hi   = lane >> 4;        // half-wave selector
    const int wv   = tid >> 5;         // wave id (0..7)

    const int i0x = sI0[n][0], i0y = sI0[n][1], i0z = sI0[n][2];
    const int base = (i0x << 16) + (i0y << 8) + i0z;   // flat idx of window start

    // B fragments (K layout: VGPR j, lane l -> K = 2*(l>>4) + j, per 4-K chunk)
    v2f B0, B1, B2;
    B0.x = sCoef[n][2*hi + 0];  B0.y = sCoef[n][2*hi + 1];
    B1.x = sCoef[n][4 + 2*hi];  B1.y = sCoef[n][5 + 2*hi];
    B2.x = sCoef[n][8 + 2*hi];  B2.y = sCoef[n][9 + 2*hi];

    // ---- tile loop: 108 M-tiles of 16 voxels, 8 waves round-robin ----
    for (int t = wv; t < NTILES; t += 8) {
        // A fragment: lane holds voxel row m = lane%16, K = 2*hi + j.
        // Select scalar inputs first (v_cndmask), then flat multiplies, so the
        // compiler does not if-convert these into exec-mask diamonds.
        const int pk = sPack[t*16 + (lane & 15)];
        const float fz = (float)(pk & 255);
        const float fy = (float)((pk >> 8) & 255);
        const float fx = (float)(pk >> 16);

        const float a0x = hi ? fy   : 1.0f;   // [1 | fy]
        const float a0y = hi ? fz   : fx;     // [fx | fz]
        const float p1  = hi ? fx   : fy;     // partner for fy
        const float qxx = hi ? 0.0f : fx;     // zeroed pad selectors
        const float qyy = hi ? 0.0f : fy;

        v2f A0, A1, A2;
        A0.x = a0x;        A0.y = a0y;
        A1.x = a0y * a0y;  A1.y = fy * p1;    // [fx^2,fy^2 | fz^2,fx*fy]
        A2.x = qxx * fz;   A2.y = qyy * fz;   // [fx*fz,fy*fz | 0,0]

        v8f acc = (v8f)(0.0f);
        acc = __builtin_amdgcn_wmma_f32_16x16x4_f32(false, A0, false, B0,
                                                    (short)0, acc, false, false);
        acc = __builtin_amdgcn_wmma_f32_16x16x4_f32(false, A1, false, B1,
                                                    (short)0, acc, false, false);
        acc = __builtin_amdgcn_wmma_f32_16x16x4_f32(false, A2, false, B2,
                                                    (short)0, acc, false, false);

        // This half-wave's 8 output rows are 8 consecutive voxel offsets.
        const int4* p4 = (const int4*)&sPack[t*16 + 8*hi];
        const int4 pa = p4[0];
        const int4 pb = p4[1];
        int pks[8];
        pks[0] = pa.x; pks[1] = pa.y; pks[2] = pa.z; pks[3] = pa.w;
        pks[4] = pb.x; pks[5] = pb.y; pks[6] = pb.z; pks[7] = pb.w;

        // D layout: VGPR r, lanes 0-15 -> M=r, lanes 16-31 -> M=r+8; N = lane%16
        #pragma unroll
        for (int r = 0; r < 8; ++r) {
            const float power = acc[r];
            if (power >= P_MIN) {                           // alpha >= 1/255
                const float alpha = __builtin_amdgcn_exp2f(power); // bare v_exp
                const int pk2 = pks[r];
                // base + pk2 == ((ix*256)+iy)*256+iz exactly (linear arith);
                // branchless bounds test: any axis <0 or >255 fails one cmp.
                const int ix = i0x + (pk2 >> 16);
                const int iy = i0y + ((pk2 >> 8) & 255);
                const int iz = i0z + (pk2 & 255);
                const bool ok = ((unsigned)(ix | iy | iz)) < 256u;
                const int   idx = ok ? (base + pk2) : 0;
                const float val = ok ? alpha : 0.0f;        // +0.0 is a no-op
                atomicAdd(&out[idx], val);
            }
        }
    }
}

// ---------------------------------------------------------------------------
extern "C" void kernel_launch(void* const* d_in, const int* in_sizes, int n_in,
                              void* d_out, int out_size, void* d_ws, size_t ws_size,
                              hipStream_t stream) {
    const float* means  = (const float*)d_in[0];
    const float* opac   = (const float*)d_in[1];
    const float* scales = (const float*)d_in[2];
    const float* rots   = (const float*)d_in[3];
    float* out = (float*)d_out;
    const int N = in_sizes[1];                 // opacities: one per gaussian

    gv_zero_fields<<<4096, 256, 0, stream>>>(out);

    const int blocks = (N + 15) / 16;
    gv_splat<<<blocks, 256, 0, stream>>>(means, opac, scales, rots, out, N);
}